// LocalAttention_72705206386816
// MI455X (gfx1250) — compile-verified
//
#include <hip/hip_runtime.h>
#include <hip/hip_bf16.h>
#include <math.h>

typedef float v2f __attribute__((ext_vector_type(2)));
typedef float v8f __attribute__((ext_vector_type(8)));

#define LATENT  1024
#define CONTEXT 64
#define WINDOW  129          // 2*CONTEXT+1
#define BATCH   32
#define SEQ     2048
#define HID     2048         // LAYERS*LATENT
#define ROWS    64           // LAYERS*BATCH

// -------------------------------------------------------------------------
// Branch-free K-loop for the f32 WMMA GEMM (wave32, 16x16 tile per wave).
// Fragment layout per CDNA5 ISA 7.12.2:
//   A 16x4 f32: lane half h covers K pair {k0+2h, k0+2h+1}  -> one b64 load
//   B 4x16 f32: same K pair at column n                      -> b64 if transB
//   C/D 16x16 f32: vgpr r -> M = r + 8*h, N = lane&15
// kofsB lets the concat caller shift the K origin inside B.
// -------------------------------------------------------------------------
template<bool TRANSB>
__device__ __forceinline__ v8f wmma_kloop(const float* __restrict__ A, int K,
                                          const float* __restrict__ B, int ldb,
                                          int kofsB, int m, int n, int half, v8f acc)
{
    const v2f* __restrict__ Ap = reinterpret_cast<const v2f*>(A + (size_t)m * K);
    const v2f* __restrict__ Bp = TRANSB
        ? reinterpret_cast<const v2f*>(B + (size_t)n * ldb + kofsB) : nullptr;

    #pragma unroll 8
    for (int k0 = 0; k0 < K; k0 += 4) {
        const int ka = k0 + 2 * half;
        v2f a = Ap[ka >> 1];
        v2f b;
        if (TRANSB) {
            b = Bp[ka >> 1];
        } else {
            const float* c0 = B + (size_t)(kofsB + ka) * ldb + n;
            b.x = c0[0];
            b.y = c0[ldb];
        }
        acc = __builtin_amdgcn_wmma_f32_16x16x4_f32(
            false, a, false, b, (short)0, acc, false, false);
    }
    return acc;
}

// D[m][n] = act( A1[m,0:K1] ++ A2[m,0:K2]  dot  B(:,n)  + bias[n] )
template<bool TRANSB, bool CONCAT, bool TANH>
__global__ void wmma_gemm(const float* __restrict__ A1, int K1,
                          const float* __restrict__ A2, int K2,
                          const float* __restrict__ Bmat, int ldb,
                          const float* __restrict__ bias,
                          float* __restrict__ D, int ldd)
{
    const int lane = threadIdx.x;      // 0..31
    const int half = lane >> 4;
    const int l16  = lane & 15;
    const int n0 = blockIdx.x * 16;
    const int m0 = blockIdx.y * 16;
    const int m  = m0 + l16;
    const int n  = n0 + l16;

    v8f acc = {};
    acc = wmma_kloop<TRANSB>(A1, K1, Bmat, ldb, 0, m, n, half, acc);
    if (CONCAT)
        acc = wmma_kloop<TRANSB>(A2, K2, Bmat, ldb, K1, m, n, half, acc);

    const float bv = bias ? bias[n0 + l16] : 0.0f;
    #pragma unroll
    for (int r = 0; r < 8; ++r) {
        const int row = m0 + r + 8 * half;
        float v = acc[r] + bv;
        if (TANH) v = tanhf(v);
        D[(size_t)row * ldd + (n0 + l16)] = v;
    }
}

// -------------------------------------------------------------------------
// Per row m (= l*B+b): pt = sigmoid(dot(T[m], W_vp) + b_vp) * SEQ,
// trunc(pt), and sc0 = dot(ht[m], b_att) (folded bias term of the scores).
// -------------------------------------------------------------------------
__global__ void predict_reduce(const float* __restrict__ T,
                               const float* __restrict__ ht,
                               const float* __restrict__ W_vp,
                               const float* __restrict__ b_vp,
                               const float* __restrict__ b_att,
                               float* __restrict__ pt_val,
                               float* __restrict__ pt_tr,
                               float* __restrict__ sc0)
{
    const int m = blockIdx.x;
    const int t = threadIdx.x;                 // 256 threads
    __shared__ float s1[256], s2[256];
    float a = 0.f, c = 0.f;
    for (int k = t; k < LATENT; k += 256) {
        a += T [m * LATENT + k] * W_vp[k];
        c += ht[m * LATENT + k] * b_att[k];
    }
    s1[t] = a; s2[t] = c;
    __syncthreads();
    for (int off = 128; off > 0; off >>= 1) {
        if (t < off) { s1[t] += s1[t + off]; s2[t] += s2[t + off]; }
        __syncthreads();
    }
    if (t == 0) {
        const float z = s1[0] + b_vp[0];
        const float p = (1.0f / (1.0f + expf(-z))) * (float)SEQ;
        pt_val[m] = p;
        pt_tr [m] = truncf(p);
        sc0  [m] = s2[0];
    }
}

// -------------------------------------------------------------------------
// Fused window gather + scores + softmax * gaussian + context vector.
// One block per (l,b). scores[w] = hs_sub[w,:].q + sc0 (zeros outside pad).
// -------------------------------------------------------------------------
__global__ void local_attn(const float* __restrict__ hs,      // [B,S,HID]
                           const float* __restrict__ q,       // [ROWS,LATENT]
                           const float* __restrict__ pt_val,
                           const float* __restrict__ pt_tr,
                           const float* __restrict__ sc0,
                           float* __restrict__ at_out,        // [ROWS,WINDOW]
                           float* __restrict__ ct)            // [ROWS,LATENT]
{
    const int m = blockIdx.x;                 // l*BATCH + b
    const int l = m >> 5, b = m & 31;
    const int t = threadIdx.x;                // 256
    const int wave = t >> 5, lane = t & 31;

    __shared__ float sc[WINDOW];
    __shared__ float at[WINDOW];
    __shared__ float red[2];

    const float ptv  = pt_val[m];
    const int   base = (int)pt_tr[m];         // trunc(pt) >= 0
    const float* qrow = q + (size_t)m * LATENT;
    const float* hsb  = hs + (size_t)b * SEQ * HID + (size_t)l * LATENT;

    // ---- scores: one wave per window position, lane-strided dot over d ----
    for (int w = wave; w < WINDOW; w += 8) {
        int idxp = w + base;                              // padded coord
        if (idxp > SEQ + 2 * CONTEXT - 1) idxp = SEQ + 2 * CONTEXT - 1;
        if (idxp < 0) idxp = 0;
        const int s = idxp - CONTEXT;
        float acc = 0.f;
        if (s >= 0 && s < SEQ) {
            const float* row = hsb + (size_t)s * HID;
            // warm L2/WGP$ for the ct pass
            __builtin_prefetch(row + lane * 8, 0, 1);
            for (int k = lane; k < LATENT; k += 32)
                acc += row[k] * qrow[k];
        }
        for (int off = 16; off > 0; off >>= 1)
            acc += __shfl_xor(acc, off, 32);
        if (lane == 0) sc[w] = acc + sc0[m];
    }
    __syncthreads();

    // ---- softmax (129 elems; serial max/sum is trivial) ----
    if (t == 0) {
        float mx = sc[0];
        for (int w = 1; w < WINDOW; ++w) mx = fmaxf(mx, sc[w]);
        red[0] = mx;
    }
    __syncthreads();
    if (t < WINDOW) at[t] = expf(sc[t] - red[0]);
    __syncthreads();
    if (t == 0) {
        float sm = 0.f;
        for (int w = 0; w < WINDOW; ++w) sm += at[w];
        red[1] = sm;
    }
    __syncthreads();
    if (t < WINDOW) {
        const float diff = ((float)(t - CONTEXT) + (float)base) - ptv;
        const float r    = diff / (float)HID;
        const float g    = expf(-2.0f * r * r);
        const float a    = (at[t] / red[1]) * g;
        at[t] = a;
        at_out[(size_t)m * WINDOW + t] = a;
    }
    __syncthreads();

    // ---- ct[d] = sum_w at[w] * hs_sub[w,d] ----
    for (int d = t; d < LATENT; d += 256) {
        float acc = 0.f;
        for (int w = 0; w < WINDOW; ++w) {
            int idxp = w + base;
            if (idxp > SEQ + 2 * CONTEXT - 1) idxp = SEQ + 2 * CONTEXT - 1;
            const int s = idxp - CONTEXT;
            if (s >= 0 && s < SEQ)
                acc += at[w] * hsb[(size_t)s * HID + d];
        }
        ct[(size_t)m * LATENT + d] = acc;
    }
}

// -------------------------------------------------------------------------
extern "C" void kernel_launch(void* const* d_in, const int* in_sizes, int n_in,
                              void* d_out, int out_size, void* d_ws, size_t ws_size,
                              hipStream_t stream)
{
    (void)in_sizes; (void)n_in; (void)out_size; (void)ws_size;
    const float* hs    = (const float*)d_in[0];
    const float* ht    = (const float*)d_in[1];   // [ROWS,LATENT] flat
    const float* W_att = (const float*)d_in[2];   // [1024,1024]
    const float* b_att = (const float*)d_in[3];
    const float* W_ctx = (const float*)d_in[4];   // [1024,2048]
    const float* b_ctx = (const float*)d_in[5];
    const float* W_vp  = (const float*)d_in[6];   // [1,1024]
    const float* b_vp  = (const float*)d_in[7];
    const float* W_pos = (const float*)d_in[8];   // [1024,1024]
    const float* b_pos = (const float*)d_in[9];

    float* out = (float*)d_out;                   // [ROWS*LATENT] out, then [ROWS*WINDOW] at
    float* ws  = (float*)d_ws;

    float* T    = ws;                 // 64*1024
    float* q    = ws + 65536;         // 64*1024
    float* ct   = ws + 131072;        // 64*1024
    float* ptv  = ws + 196608;        // 64
    float* ptt  = ws + 196672;        // 64
    float* sc0  = ws + 196736;        // 64

    dim3 gTiles(LATENT / 16, ROWS / 16);          // (64, 4)
    dim3 blk(32);

    // T = tanh(ht @ W_pos^T + b_pos)      (B transposed, tanh)
    wmma_gemm<true, false, true><<<gTiles, blk, 0, stream>>>(
        ht, LATENT, ht, 0, W_pos, LATENT, b_pos, T, LATENT);
    // q = ht @ W_att                      (B direct, no bias/act)
    wmma_gemm<false, false, false><<<gTiles, blk, 0, stream>>>(
        ht, LATENT, ht, 0, W_att, LATENT, nullptr, q, LATENT);
    // pt / trunc(pt) / sc0
    predict_reduce<<<ROWS, 256, 0, stream>>>(T, ht, W_vp, b_vp, b_att, ptv, ptt, sc0);
    // scores + softmax*gaussian -> at (output) and ct
    local_attn<<<ROWS, 256, 0, stream>>>(hs, q, ptv, ptt, sc0,
                                         out + ROWS * LATENT, ct);
    // out = tanh(concat(ct, ht) @ W_ctx^T + b_ctx)
    wmma_gemm<true, true, true><<<gTiles, blk, 0, stream>>>(
        ct, LATENT, ht, LATENT, W_ctx, 2 * LATENT, b_ctx, out, LATENT);
}